// NVGPTDecoderLayer_5111011082472
// MI455X (gfx1250) — compile-verified
//
#include <hip/hip_runtime.h>
#include <hip/hip_bf16.h>
#include <math.h>

// Problem constants (from reference)
#define HIDN 4096
#define BATCH 8
#define SEQ 4096
#define NHEAD 32
#define HDIM 128
#define FFN_DIM 16384
#define EPSV 1e-5f
#define ATT_SCALE 0.08838834764831845f  // 128^-0.5

typedef float v2f __attribute__((ext_vector_type(2)));
typedef float v4f __attribute__((ext_vector_type(4)));
typedef float v8f __attribute__((ext_vector_type(8)));

// ---------------------------------------------------------------------------
// LayerNorm with (w + 1) scale:  out = (x - mu) * rsqrt(var + eps) * (w+1) + b
// One block per row (8 rows), 256 threads.
// ---------------------------------------------------------------------------
__global__ void ln1p_kernel(const float* __restrict__ in,
                            const float* __restrict__ w,
                            const float* __restrict__ b,
                            float* __restrict__ out) {
  __shared__ float s_sum[256];
  __shared__ float s_sq[256];
  const int row = blockIdx.x;
  const int tid = threadIdx.x;
  const float* x = in + (size_t)row * HIDN;

  float s = 0.f, ss = 0.f;
  for (int i = tid; i < HIDN; i += 256) {
    float v = x[i];
    s += v;
    ss += v * v;
  }
  s_sum[tid] = s;
  s_sq[tid] = ss;
  __syncthreads();
  for (int off = 128; off > 0; off >>= 1) {
    if (tid < off) {
      s_sum[tid] += s_sum[tid + off];
      s_sq[tid] += s_sq[tid + off];
    }
    __syncthreads();
  }
  const float mu = s_sum[0] * (1.f / HIDN);
  const float var = s_sq[0] * (1.f / HIDN) - mu * mu;
  const float r = rsqrtf(var + EPSV);
  for (int i = tid; i < HIDN; i += 256) {
    out[(size_t)row * HIDN + i] = (x[i] - mu) * r * (w[i] + 1.f) + b[i];
  }
}

// ---------------------------------------------------------------------------
// Skinny GEMM with fp32 WMMA:  out[8,N] = A[8,K] @ W[K,N]  (+ resid[8,N])
// M padded 8->16 with zero rows via lane masking.
// Block = 128 threads = 4 waves; each wave owns one 16-column tile.
// A 16x4 f32 frag: lane = M row; lane-half selects K pair; VGPR0/1 = K, K+1.
// B 4x16  f32 frag: mirrored (lane = N col; lane-half selects K pair).
// C/D 16x16 f32: acc[r] for lanes 0-15 holds row M=r  -> only those store.
// W is streamed exactly once -> non-temporal loads keep it out of L2 so the
// activation matrix A (read by every wave) stays resident. Two independent
// accumulators break the WMMA D->C dependency chain for XDL ILP.
// ---------------------------------------------------------------------------
__global__ void gemm_wmma_f32_kernel(const float* __restrict__ A,
                                     const float* __restrict__ W,
                                     const float* __restrict__ resid,
                                     float* __restrict__ out,
                                     int K, int N) {
  const int lane = threadIdx.x & 31;
  const int wave = threadIdx.x >> 5;
  const int n0 = blockIdx.x * 64 + wave * 16;
  const int half = lane >> 4;     // 0: K=k0..k0+1,  1: K=k0+2..k0+3
  const int l = lane & 15;
  const float amask = (l < 8) ? 1.f : 0.f;   // pad M rows 8..15 with zeros
  const int arow = (l < 8) ? l : 0;

  const float* Arow = A + (size_t)arow * K;
  const size_t bcol = (size_t)n0 + l;

  v8f acc0 = {0.f, 0.f, 0.f, 0.f, 0.f, 0.f, 0.f, 0.f};
  v8f acc1 = {0.f, 0.f, 0.f, 0.f, 0.f, 0.f, 0.f, 0.f};

  for (int k0 = 0; k0 < K; k0 += 8) {
    const int ka = k0 + half * 2;
    v2f a0, a1, b0, b1;
    a0.x = amask * Arow[ka];
    a0.y = amask * Arow[ka + 1];
    a1.x = amask * Arow[ka + 4];
    a1.y = amask * Arow[ka + 5];
    b0.x = __builtin_nontemporal_load(&W[(size_t)ka * N + bcol]);
    b0.y = __builtin_nontemporal_load(&W[(size_t)(ka + 1) * N + bcol]);
    b1.x = __builtin_nontemporal_load(&W[(size_t)(ka + 4) * N + bcol]);
    b1.y = __builtin_nontemporal_load(&W[(size_t)(ka + 5) * N + bcol]);
    // 8-arg form: (neg_a, A, neg_b, B, c_mod, C, reuse_a, reuse_b)
    acc0 = __builtin_amdgcn_wmma_f32_16x16x4_f32(
        false, a0, false, b0, (short)0, acc0, false, false);
    acc1 = __builtin_amdgcn_wmma_f32_16x16x4_f32(
        false, a1, false, b1, (short)0, acc1, false, false);
  }
  const v8f acc = acc0 + acc1;

  if (half == 0) {   // lanes 0-15 hold rows 0..7 in acc[0..7]
#pragma unroll
    for (int r = 0; r < 8; ++r) {
      const size_t oi = (size_t)r * N + n0 + l;
      float v = acc[r];
      if (resid) v += resid[oi];
      out[oi] = v;
    }
  }
}

// ---------------------------------------------------------------------------
// RoPE on the first 64 dims of q and k heads. qkv = [8, 3*4096] packed.
// grid = B*NH blocks, 128 threads (one per head dim).
// ---------------------------------------------------------------------------
__global__ void rope_kernel(const float* __restrict__ qkv,
                            const int* __restrict__ positions,
                            float* __restrict__ q_r,
                            float* __restrict__ k_r) {
  const int bh = blockIdx.x;           // b*32 + h
  const int b = bh >> 5;
  const int h = bh & 31;
  const int d = threadIdx.x;           // 0..127
  const float p = (float)positions[b];

  const float* q = qkv + (size_t)b * (3 * HIDN) + (size_t)h * HDIM;
  const float* k = q + HIDN;

  float qo, ko;
  if (d < 64) {
    const int i = (d < 32) ? d : d - 32;
    const float inv_freq = powf(10000.f, -(float)i / 32.f);
    const float ang = p * inv_freq;
    const float c = cosf(ang);
    const float s = sinf(ang);
    const float x1q = q[i], x2q = q[i + 32];
    const float x1k = k[i], x2k = k[i + 32];
    if (d < 32) {
      qo = x1q * c - x2q * s;
      ko = x1k * c - x2k * s;
    } else {
      qo = x2q * c + x1q * s;
      ko = x2k * c + x1k * s;
    }
  } else {
    qo = q[d];
    ko = k[d];
  }
  q_r[(size_t)bh * HDIM + d] = qo;
  k_r[(size_t)bh * HDIM + d] = ko;
}

// ---------------------------------------------------------------------------
// Flash-decode attention. One block per (b,h); 8 waves; each wave strides the
// causal prefix s = wave, wave+8, ... <= pos. The s == pos row uses the fresh
// roped k / new v (caches are read-only inputs and must not be mutated).
// Per s: lane loads 4 contiguous floats of K row (coalesced 512B/wave, NT hint
// since cache rows are single-use), cross-lane dot via shfl_xor, online-softmax
// accumulate of V. Waves merged via LDS at the end.
// ---------------------------------------------------------------------------
__global__ void attn_kernel(const float* __restrict__ q_r,
                            const float* __restrict__ k_r,
                            const float* __restrict__ qkv,
                            const float* __restrict__ k_cache,
                            const float* __restrict__ v_cache,
                            const int* __restrict__ positions,
                            float* __restrict__ out) {
  const int bh = blockIdx.x;        // b*32 + h
  const int b = bh >> 5;
  const int h = bh & 31;
  const int tid = threadIdx.x;      // 256
  const int wave = tid >> 5;
  const int lane = tid & 31;
  const int pos = positions[b];

  const v4f qv = *(const v4f*)(q_r + (size_t)bh * HDIM + lane * 4);

  float m = -INFINITY, lsum = 0.f;
  float a0 = 0.f, a1 = 0.f, a2 = 0.f, a3 = 0.f;

  const size_t base_b = (size_t)b * SEQ * NHEAD * HDIM;
  const float* k_new = k_r + (size_t)bh * HDIM;
  const float* v_new = qkv + (size_t)b * (3 * HIDN) + 2 * HIDN + (size_t)h * HDIM;

  for (int s = wave; s <= pos; s += 8) {
    const float* krow;
    const float* vrow;
    if (s == pos) {
      krow = k_new;
      vrow = v_new;
    } else {
      const size_t off = base_b + ((size_t)s * NHEAD + h) * HDIM;
      krow = k_cache + off;
      vrow = v_cache + off;
    }
    const v4f kv = __builtin_nontemporal_load((const v4f*)(krow + lane * 4));
    float dotp = qv.x * kv.x + qv.y * kv.y + qv.z * kv.z + qv.w * kv.w;
#pragma unroll
    for (int o = 16; o > 0; o >>= 1) dotp += __shfl_xor(dotp, o);
    const float score = dotp * ATT_SCALE;

    const v4f vv = __builtin_nontemporal_load((const v4f*)(vrow + lane * 4));
    const float mn = fmaxf(m, score);
    const float corr = expf(m - mn);
    const float pr = expf(score - mn);
    lsum = lsum * corr + pr;
    a0 = a0 * corr + pr * vv.x;
    a1 = a1 * corr + pr * vv.y;
    a2 = a2 * corr + pr * vv.z;
    a3 = a3 * corr + pr * vv.w;
    m = mn;
  }

  __shared__ float sm[8];
  __shared__ float sl[8];
  __shared__ float sacc[8][HDIM];
  sacc[wave][lane * 4 + 0] = a0;
  sacc[wave][lane * 4 + 1] = a1;
  sacc[wave][lane * 4 + 2] = a2;
  sacc[wave][lane * 4 + 3] = a3;
  if (lane == 0) {
    sm[wave] = m;
    sl[wave] = lsum;
  }
  __syncthreads();

  if (tid < HDIM) {
    float M = -INFINITY;
#pragma unroll
    for (int w = 0; w < 8; ++w) M = fmaxf(M, sm[w]);
    float L = 0.f, o = 0.f;
#pragma unroll
    for (int w = 0; w < 8; ++w) {
      const float f = expf(sm[w] - M);
      L += f * sl[w];
      o += f * sacc[w][tid];
    }
    out[(size_t)bh * HDIM + tid] = o / L;
  }
}

// ---------------------------------------------------------------------------
// SwiGLU: g = silu(h1) * h2   where h = [8, 2*FFN], h1 = h[:, :FFN]
// ---------------------------------------------------------------------------
__global__ void swiglu_kernel(const float* __restrict__ h, float* __restrict__ g) {
  const int idx = blockIdx.x * blockDim.x + threadIdx.x;  // 8*16384 total
  const int row = idx >> 14;
  const int col = idx & (FFN_DIM - 1);
  const float a = __builtin_nontemporal_load(&h[(size_t)row * (2 * FFN_DIM) + col]);
  const float c = __builtin_nontemporal_load(&h[(size_t)row * (2 * FFN_DIM) + FFN_DIM + col]);
  const float sig = 1.f / (1.f + expf(-a));
  g[idx] = a * sig * c;
}

// ---------------------------------------------------------------------------
extern "C" void kernel_launch(void* const* d_in, const int* in_sizes, int n_in,
                              void* d_out, int out_size, void* d_ws, size_t ws_size,
                              hipStream_t stream) {
  const float* hidden   = (const float*)d_in[0];
  const float* k_cache  = (const float*)d_in[1];
  const float* v_cache  = (const float*)d_in[2];
  const float* w_qkv    = (const float*)d_in[3];
  const float* w_dense  = (const float*)d_in[4];
  const float* w_fc1    = (const float*)d_in[5];
  const float* w_fc2    = (const float*)d_in[6];
  const float* ln1_w    = (const float*)d_in[7];
  const float* ln1_b    = (const float*)d_in[8];
  const float* ln2_w    = (const float*)d_in[9];
  const float* ln2_b    = (const float*)d_in[10];
  const int*   positions= (const int*)d_in[11];
  float* out = (float*)d_out;

  float* ws  = (float*)d_ws;
  float* x1   = ws;              // 8*4096   = 32768
  float* qkv  = x1  + 32768;     // 8*12288  = 98304
  float* q_r  = qkv + 98304;     // 8*4096
  float* k_r  = q_r + 32768;     // 8*4096
  float* attn = k_r + 32768;     // 8*4096
  float* hs2  = attn+ 32768;     // 8*4096
  float* x2   = hs2 + 32768;     // 8*4096
  float* hbuf = x2  + 32768;     // 8*32768  = 262144
  float* g    = hbuf+ 262144;    // 8*16384  = 131072   (total ~2.75 MB)

  // 1) LN1
  ln1p_kernel<<<8, 256, 0, stream>>>(hidden, ln1_w, ln1_b, x1);
  // 2) QKV projection: [8,4096] @ [4096,12288]
  gemm_wmma_f32_kernel<<<12288 / 64, 128, 0, stream>>>(x1, w_qkv, nullptr, qkv, HIDN, 3 * HIDN);
  // 3) RoPE on q, k
  rope_kernel<<<BATCH * NHEAD, 128, 0, stream>>>(qkv, positions, q_r, k_r);
  // 4) Flash-decode attention (reads caches, substitutes fresh k/v at s==pos)
  attn_kernel<<<BATCH * NHEAD, 256, 0, stream>>>(q_r, k_r, qkv, k_cache, v_cache,
                                                 positions, attn);
  // 5) Dense + residual: [8,4096] @ [4096,4096] + hidden
  gemm_wmma_f32_kernel<<<4096 / 64, 128, 0, stream>>>(attn, w_dense, hidden, hs2, HIDN, HIDN);
  // 6) LN2
  ln1p_kernel<<<8, 256, 0, stream>>>(hs2, ln2_w, ln2_b, x2);
  // 7) FC1: [8,4096] @ [4096,32768]
  gemm_wmma_f32_kernel<<<32768 / 64, 128, 0, stream>>>(x2, w_fc1, nullptr, hbuf, HIDN, 2 * FFN_DIM);
  // 8) SwiGLU
  swiglu_kernel<<<(BATCH * FFN_DIM) / 256, 256, 0, stream>>>(hbuf, g);
  // 9) FC2 + residual -> output: [8,16384] @ [16384,4096] + hs2
  gemm_wmma_f32_kernel<<<4096 / 64, 128, 0, stream>>>(g, w_fc2, hs2, out, FFN_DIM, HIDN);
}